// Float32HardwareSurrogate_52347061404309
// MI455X (gfx1250) — compile-verified
//
#include <hip/hip_runtime.h>

typedef float v2f __attribute__((ext_vector_type(2)));
typedef float v8f __attribute__((ext_vector_type(8)));
typedef int   v4i __attribute__((ext_vector_type(4)));

#define INV_HW_SCALE (1.0f / 64.0f)
#define NITER 8            // image-groups (16 imgs each) per wave
#define WAVES_PER_BLOCK 4
// floats per wave-private LDS region: 2x1024 (double-buffered 16 imgs) + 256 (conv) + 64 (obuf)
#define WAVE_LDS (2 * 1024 + 256 + 64)

__device__ __forceinline__ float clip128(float x) {
    return fminf(fmaxf(x, -128.0f), 127.0f);
}

// One wave-wide 512B copy: 32 lanes x 16B, global -> LDS.
__device__ __forceinline__ void stage512(const float4* __restrict__ g, float4* l, int lane) {
#if __has_builtin(__builtin_amdgcn_global_load_async_to_lds_b128)
    __builtin_amdgcn_global_load_async_to_lds_b128(
        (__attribute__((address_space(1))) v4i*)(g + lane),
        (__attribute__((address_space(3))) v4i*)(l + lane),
        0, 0);
#else
    l[lane] = g[lane];
#endif
}

__device__ __forceinline__ void wait_stage_keep8() {
#if __has_builtin(__builtin_amdgcn_s_wait_asynccnt)
    __builtin_amdgcn_s_wait_asynccnt(8);
#else
    asm volatile("s_wait_asynccnt 8" ::: "memory");
#endif
    asm volatile("" ::: "memory");
}

__device__ __forceinline__ void wait_stage_all() {
#if __has_builtin(__builtin_amdgcn_s_wait_asynccnt)
    __builtin_amdgcn_s_wait_asynccnt(0);
#else
    asm volatile("s_wait_asynccnt 0" ::: "memory");
#endif
    asm volatile("" ::: "memory");
}

__global__ __launch_bounds__(WAVES_PER_BLOCK * 32)
void Float32HardwareSurrogate_52347061404309_kernel(
    const float* __restrict__ img1, const float* __restrict__ img2,
    const float* __restrict__ Wc,   const float* __restrict__ Wl,
    float* __restrict__ out, int groupsPerInput, long long nImgs)
{
    __shared__ float lds[WAVES_PER_BLOCK * WAVE_LDS];

    const int lane = threadIdx.x & 31;
    const int wave = threadIdx.x >> 5;

    float* buf0 = &lds[wave * WAVE_LDS];   // 16 images x 64 floats
    float* buf1 = buf0 + 1024;             // second staging buffer
    float* conv = buf0 + 2048;             // 16 images x 16 features
    float* obuf = conv + 256;              // 16 images x 4 outputs

    const long long totalGroups = 2LL * groupsPerInput;
    const long long base = ((long long)blockIdx.x * WAVES_PER_BLOCK + wave) * NITER;
    if (base >= totalGroups) return;       // wave-uniform exit

    const int half = lane >> 4;            // which half-wave
    const int m    = lane & 15;            // A-row within tile / B,C,D column
    const int n    = m;

    // ---- Per-lane B fragments for W_conv / W_linear, K-chunks k=0..3 ----
    // B layout (4x16, KxN): v0: K=4k+2*half, v1: K=4k+2*half+1, N = lane&15.
    v2f bc[4], bl[4];
    #pragma unroll
    for (int k = 0; k < 4; ++k) {
        #pragma unroll
        for (int j = 0; j < 2; ++j) {
            int kk = 4 * k + 2 * half + j;
            bc[k][j] = (n < 4) ? Wc[kk * 4 + n] : 0.0f;
            bl[k][j] = (n < 4) ? Wl[kk * 4 + n] : 0.0f;
        }
    }

    // Issue the full 4KB staging (8 x 512B async copies) for combined group c.
    auto issue = [&](long long c, float* buf) {
        const float* s = (c < groupsPerInput) ? (img1 + c * 1024)
                                              : (img2 + (c - groupsPerInput) * 1024);
        const float4* g4 = (const float4*)s;
        float4*       l4 = (float4*)buf;
        #pragma unroll
        for (int i = 0; i < 8; ++i)                 // 256 float4s = 16 images
            stage512(g4 + i * 32, l4 + i * 32, lane);
    };

    issue(base, buf0);

    for (int it = 0; it < NITER; ++it) {
        const long long c = base + it;
        if (c >= totalGroups) break;                // wave-uniform

        float* imgs = (it & 1) ? buf1 : buf0;
        const bool pf = (it + 1 < NITER) && (c + 1 < totalGroups);
        if (pf) issue(c + 1, (it & 1) ? buf0 : buf1);

        // Async loads complete in order: <=8 outstanding means current buffer landed.
        if (pf) wait_stage_keep8();
        else    wait_stage_all();

        // ---- Matmul 1: tile t covers images 4t..4t+3; A(16x16) @ Wc(16x4) ----
        // A row M = imgInTile*4 + patch; K = flattened 4x4 patch element.
        v8f acc1[4];
        #pragma unroll
        for (int t = 0; t < 4; ++t) {
            v8f cacc = {};
            #pragma unroll
            for (int k = 0; k < 4; ++k) {
                v2f a;
                #pragma unroll
                for (int j = 0; j < 2; ++j) {
                    int e   = 4 * k + 2 * half + j;   // K index (patch element)
                    int it4 = m >> 2;                 // image within tile
                    int p   = m & 3;                  // patch index
                    int row = ((p >> 1) << 2) + (e >> 2);
                    int col = ((p & 1) << 2) + (e & 3);
                    a[j] = imgs[(t * 4 + it4) * 64 + row * 8 + col];
                }
                cacc = __builtin_amdgcn_wmma_f32_16x16x4_f32(
                           false, a, false, bc[k], (short)0, cacc, false, false);
            }
            acc1[t] = cacc;
        }

        // ---- relu -> /64 -> clip; scatter to conv[img][patch*4+ch] ----
        // D layout: VGPR r: lanes0-15 -> M=r, lanes16-31 -> M=r+8; N = lane&15.
        #pragma unroll
        for (int t = 0; t < 4; ++t) {
            if (n < 4) {
                #pragma unroll
                for (int r = 0; r < 8; ++r) {
                    int   M = r + 8 * half;
                    float v = fmaxf(acc1[t][r], 0.0f) * INV_HW_SCALE;
                    v = clip128(v);
                    int imgLocal = t * 4 + (M >> 2);
                    int f        = (M & 3) * 4 + n;
                    conv[imgLocal * 16 + f] = v;
                }
            }
        }

        // ---- Matmul 2: conv(16x16) @ Wl(16x4); A row M = image ----
        v8f c2 = {};
        #pragma unroll
        for (int k = 0; k < 4; ++k) {
            v2f a;
            #pragma unroll
            for (int j = 0; j < 2; ++j) {
                int e = 4 * k + 2 * half + j;
                a[j] = conv[m * 16 + e];
            }
            c2 = __builtin_amdgcn_wmma_f32_16x16x4_f32(
                     false, a, false, bl[k], (short)0, c2, false, false);
        }

        // ---- /64 -> clip; stage through LDS, then coalesced float4 stores ----
        if (n < 4) {
            #pragma unroll
            for (int r = 0; r < 8; ++r) {
                int M = r + 8 * half;
                obuf[M * 4 + n] = clip128(c2[r] * INV_HW_SCALE);
            }
        }
        float* dst = (c < groupsPerInput)
                   ? (out + c * 64)
                   : (out + nImgs * 4 + (c - groupsPerInput) * 64);
        if (lane < 16) {
            float4 o = ((const float4*)obuf)[lane];
            ((float4*)dst)[lane] = o;
        }
    }
}

extern "C" void kernel_launch(void* const* d_in, const int* in_sizes, int n_in,
                              void* d_out, int out_size, void* d_ws, size_t ws_size,
                              hipStream_t stream) {
    const float* img1 = (const float*)d_in[0];
    const float* img2 = (const float*)d_in[1];
    const float* Wc   = (const float*)d_in[2];
    const float* Wl   = (const float*)d_in[3];
    float* out = (float*)d_out;

    long long nImgs = (long long)in_sizes[0] / 64;          // 8x8 images per input
    int groupsPerInput = (int)(nImgs / 16);                 // 16 images per group
    long long totalGroups = 2LL * groupsPerInput;
    long long wavesNeeded = (totalGroups + NITER - 1) / NITER;
    int blocks = (int)((wavesNeeded + WAVES_PER_BLOCK - 1) / WAVES_PER_BLOCK);

    Float32HardwareSurrogate_52347061404309_kernel<<<blocks, WAVES_PER_BLOCK * 32, 0, stream>>>(
        img1, img2, Wc, Wl, out, groupsPerInput, nImgs);
}